// NetGCN_17188459118902
// MI455X (gfx1250) — compile-verified
//
#include <hip/hip_runtime.h>
#include <cmath>

typedef __attribute__((ext_vector_type(2))) float v2f;
typedef __attribute__((ext_vector_type(8))) float v8f;

#define NUM_FEAT 128
#define DIM      64
#define OUT_SIZE 10
#define N_GRAPHS 128

// ---------------------------------------------------------------------------
// GEMM: C[N x 64] = op(A)[N x K] @ B[K x 64], op = identity or ReLU.
// One workgroup (4 wave32) handles TILES consecutive 16-row tiles; wave w owns
// columns [16w, 16w+16).  Each wave keeps TILES independent v8f accumulators,
// so each B-fragment load feeds TILES back-to-back independent
// V_WMMA_F32_16X16X4_F32 ops (deep XDL pipeline, B traffic cut TILES x).
// A-frag (16x4 f32): lane l -> row m = l&15, K-offset koff = 2*(l>>4), 2 VGPRs.
// B-frag (4x16 f32): lane l -> col n = l&15, same koff.
// C/D (16x16 f32, 8 VGPRs): VGPR r -> row r + 8*(l>>4), col l&15.
// N = 100000 = 16 * 6250 = 16 * 5 * 1250, so TILES=5 divides exactly:
// no bounds checks, EXEC stays all-1s (WMMA requirement).
// ---------------------------------------------------------------------------
template <int K, bool RELU, int TILES>
__global__ __launch_bounds__(128) void gemm_wmma_f32(
    const float* __restrict__ A, const float* __restrict__ B,
    float* __restrict__ C) {
  constexpr int LDSK = K + 4;         // pad: row*(K+4) % 64 distinct per row
  constexpr int ROWS = TILES * 16;
  __shared__ float As[ROWS * LDSK];
  const int r0  = blockIdx.x * ROWS;
  const int tid = threadIdx.x;

  // Cooperative float4 load of the ROWS x K A-tile.
  constexpr int NV4 = ROWS * K / 4;
  for (int i = tid; i < NV4; i += 128) {
    const int m  = i / (K / 4);
    const int c4 = (i % (K / 4)) * 4;
    float4 v = *(const float4*)(A + (size_t)(r0 + m) * K + c4);
    if (RELU) {
      v.x = fmaxf(v.x, 0.0f); v.y = fmaxf(v.y, 0.0f);
      v.z = fmaxf(v.z, 0.0f); v.w = fmaxf(v.w, 0.0f);
    }
    *(float4*)(&As[m * LDSK + c4]) = v;
  }
  __syncthreads();

  const int lane    = tid & 31;
  const int wave    = tid >> 5;
  const int colbase = wave * 16;
  const int mn      = lane & 15;      // A row-in-tile / B+C col for this lane
  const int koff    = (lane >> 4) * 2;

  v8f acc[TILES];
#pragma unroll
  for (int t = 0; t < TILES; ++t) acc[t] = (v8f){0.f,0.f,0.f,0.f,0.f,0.f,0.f,0.f};

#pragma unroll
  for (int kk = 0; kk < K; kk += 4) {
    v2f b;
    b[0] = B[(kk + koff)     * DIM + colbase + mn];
    b[1] = B[(kk + koff + 1) * DIM + colbase + mn];
#pragma unroll
    for (int t = 0; t < TILES; ++t) {
      v2f a;
      a[0] = As[(t * 16 + mn) * LDSK + kk + koff];
      a[1] = As[(t * 16 + mn) * LDSK + kk + koff + 1];
      acc[t] = __builtin_amdgcn_wmma_f32_16x16x4_f32(
          /*neg_a=*/false, a, /*neg_b=*/false, b,
          /*c_mod=*/(short)0, acc[t], /*reuse_a=*/false, /*reuse_b=*/false);
    }
  }

  const int rowoff = (lane >> 4) * 8;
#pragma unroll
  for (int t = 0; t < TILES; ++t)
#pragma unroll
    for (int r = 0; r < 8; ++r)
      C[(size_t)(r0 + t * 16 + rowoff + r) * DIM + colbase + mn] = acc[t][r];
}

// ---------------------------------------------------------------------------
// Edge message-passing: out[dst[e]][:] += h[src[e]][:].
// One thread per (edge, 4-dim chunk): float4 coalesced read + 4 HW f32 atomics.
// ---------------------------------------------------------------------------
__global__ __launch_bounds__(256) void edge_scatter_add(
    const float* __restrict__ h, const int* __restrict__ src,
    const int* __restrict__ dst, float* __restrict__ out, int nE) {
  const long long t = (long long)blockIdx.x * blockDim.x + threadIdx.x;
  const int e = (int)(t >> 4);
  if (e >= nE) return;
  const int d4 = (int)(t & 15) * 4;
  const int s = src[e], d = dst[e];
  const float4 v = *(const float4*)(h + (size_t)s * DIM + d4);
  float* o = out + (size_t)d * DIM + d4;
  unsafeAtomicAdd(o + 0, v.x);
  unsafeAtomicAdd(o + 1, v.y);
  unsafeAtomicAdd(o + 2, v.z);
  unsafeAtomicAdd(o + 3, v.w);
}

// ---------------------------------------------------------------------------
// Segment sum for mean-pool: sums[batch[n]][:] += h[n][:], counts[g] += 1.
// ---------------------------------------------------------------------------
__global__ __launch_bounds__(256) void pool_sum(
    const float* __restrict__ h, const int* __restrict__ batch,
    float* __restrict__ sums, float* __restrict__ counts, int nN) {
  const long long t = (long long)blockIdx.x * blockDim.x + threadIdx.x;
  const int n = (int)(t >> 4);
  if (n >= nN) return;
  const int d4 = (int)(t & 15) * 4;
  const int g = batch[n];
  const float4 v = *(const float4*)(h + (size_t)n * DIM + d4);
  float* o = sums + (size_t)g * DIM + d4;
  unsafeAtomicAdd(o + 0, v.x);
  unsafeAtomicAdd(o + 1, v.y);
  unsafeAtomicAdd(o + 2, v.z);
  unsafeAtomicAdd(o + 3, v.w);
  if (d4 == 0) unsafeAtomicAdd(&counts[g], 1.0f);
}

// ---------------------------------------------------------------------------
// Head: out[g][o] = tanh( (sums[g]/max(cnt,1)) . Wfc[:,o] ).  1280 outputs.
// ---------------------------------------------------------------------------
__global__ __launch_bounds__(128) void head_fc_tanh(
    const float* __restrict__ sums, const float* __restrict__ counts,
    const float* __restrict__ Wfc, float* __restrict__ out) {
  const int t = blockIdx.x * blockDim.x + threadIdx.x;
  if (t >= N_GRAPHS * OUT_SIZE) return;
  const int g = t / OUT_SIZE, o = t % OUT_SIZE;
  const float inv = 1.0f / fmaxf(counts[g], 1.0f);
  float acc = 0.0f;
#pragma unroll
  for (int k = 0; k < DIM; ++k)
    acc += sums[g * DIM + k] * Wfc[k * OUT_SIZE + o];
  out[t] = tanhf(acc * inv);
}

// ---------------------------------------------------------------------------
extern "C" void kernel_launch(void* const* d_in, const int* in_sizes, int n_in,
                              void* d_out, int out_size, void* d_ws,
                              size_t ws_size, hipStream_t stream) {
  const float* x     = (const float*)d_in[0];
  const int*   ei    = (const int*)d_in[1];
  const int*   batch = (const int*)d_in[2];
  const float* W1    = (const float*)d_in[3];
  const float* W2    = (const float*)d_in[4];
  const float* Wfc   = (const float*)d_in[5];
  float*       out   = (float*)d_out;

  const int nN = in_sizes[0] / NUM_FEAT;  // 100000 (= 16 * 5 * 1250)
  const int nE = in_sizes[1] / 2;         // 3200000
  const int* src = ei;                    // edge_index[0]
  const int* dst = ei + nE;               // edge_index[1]

  // Workspace layout (two big buffers reused across layers):
  //   hW  [nN*64]  : x@W1, then relu(agg)@W2
  //   agg [nN*64]  : layer-1 aggregation, then layer-2 aggregation
  //   sums[128*64], counts[128]
  char* ws = (char*)d_ws;
  const size_t nvBytes = (size_t)nN * DIM * sizeof(float);
  float* hW     = (float*)(ws);
  float* agg    = (float*)(ws + nvBytes);
  float* sums   = (float*)(ws + 2 * nvBytes);
  float* counts = (float*)(ws + 2 * nvBytes + (size_t)N_GRAPHS * DIM * sizeof(float));

  constexpr int TILES = 5;                 // 6250 row-tiles = 5 * 1250 blocks
  const int gemmBlocks = nN / (16 * TILES);
  const long long edgeT = (long long)nE * 16;
  const int edgeBlocks  = (int)((edgeT + 255) / 256);
  const long long nodeT = (long long)nN * 16;
  const int nodeBlocks  = (int)((nodeT + 255) / 256);

  // Zero agg + sums + counts (contiguous region), graph-capture safe.
  hipMemsetAsync(agg, 0,
                 nvBytes + (size_t)N_GRAPHS * DIM * sizeof(float) +
                     (size_t)N_GRAPHS * sizeof(float),
                 stream);

  // Layer 1: hW = x @ W1 ; agg = scatter_add(hW over edges)
  gemm_wmma_f32<NUM_FEAT, false, TILES><<<gemmBlocks, 128, 0, stream>>>(x, W1, hW);
  edge_scatter_add<<<edgeBlocks, 256, 0, stream>>>(hW, src, dst, agg, nE);

  // Layer 2: hW = relu(agg) @ W2 ; re-zero agg ; agg = scatter_add(hW)
  gemm_wmma_f32<DIM, true, TILES><<<gemmBlocks, 128, 0, stream>>>(agg, W2, hW);
  hipMemsetAsync(agg, 0, nvBytes, stream);
  edge_scatter_add<<<edgeBlocks, 256, 0, stream>>>(hW, src, dst, agg, nE);

  // Mean pool + FC head + tanh
  pool_sum<<<nodeBlocks, 256, 0, stream>>>(agg, batch, sums, counts, nN);
  head_fc_tanh<<<(N_GRAPHS * OUT_SIZE + 127) / 128, 128, 0, stream>>>(
      sums, counts, Wfc, out);
}